// TranConv_81836306858498
// MI455X (gfx1250) — compile-verified
//
#include <hip/hip_runtime.h>
#include <math.h>

// Problem constants (match reference): H=128 channels, ED=32 edge feature dim.
#define H   128
#define ED  32
#define LDS_STRIDE 132  // 128 + 4 pad to avoid 16-way LDS bank conflicts on A-frag reads

typedef float v2f  __attribute__((ext_vector_type(2)));
typedef float v8f  __attribute__((ext_vector_type(8)));
typedef unsigned u32x4 __attribute__((ext_vector_type(4)));
typedef int   i32x4 __attribute__((ext_vector_type(4)));
typedef int   i32x8 __attribute__((ext_vector_type(8)));

// TDM (tensor_load_to_lds) availability: guarded so host pass / older toolchains fall back.
#if defined(__HIP_DEVICE_COMPILE__) && __has_builtin(__builtin_amdgcn_tensor_load_to_lds) && \
    __has_builtin(__builtin_amdgcn_s_wait_tensorcnt)
#define USE_TDM 1
#else
#define USE_TDM 0
#endif

// ---- ordered-uint encoding so atomicMax(uint) implements float max (incl. negatives)
__device__ __forceinline__ unsigned fenc(float f) {
  unsigned u = __float_as_uint(f);
  return (u & 0x80000000u) ? ~u : (u | 0x80000000u);
}
__device__ __forceinline__ float fdec(unsigned u) {
  unsigned v = (u & 0x80000000u) ? (u ^ 0x80000000u) : ~u;
  return __uint_as_float(v);
}
__device__ __forceinline__ void atomAddF(float* p, float v) {
  // relaxed agent-scope fadd -> global_atomic_add_f32 (returnless)
  __hip_atomic_fetch_add(p, v, __ATOMIC_RELAXED, __HIP_MEMORY_SCOPE_AGENT);
}

// =====================================================================
// Fused 4-weight node projection: O_i = A @ W_i + b_i   (A: M x H, W_i: H x H)
// One block = one 16-row M tile. The tile is DMA'd into LDS by the Tensor
// Data Mover (TENSOR_LOAD_TO_LDS) with hardware row padding reproducing the
// 132-float LDS stride (pad 4 DWORDs every 128 DWORDs); tensor_dim1 = M-m0
// gives hardware OOB-zeroing for partial tiles. All 8 waves then run
// 4 weights x 32 K-steps of V_WMMA_F32_16X16X4_F32 (fp32-exact).
// =====================================================================
__global__ __launch_bounds__(256) void gemm4_kernel(
    const float* __restrict__ A, int M,
    const float* __restrict__ W0, const float* __restrict__ b0, float* __restrict__ O0,
    const float* __restrict__ W1, const float* __restrict__ b1, float* __restrict__ O1,
    const float* __restrict__ W2, const float* __restrict__ b2, float* __restrict__ O2,
    const float* __restrict__ W3, const float* __restrict__ b3, float* __restrict__ O3)
{
  __shared__ float As[16 * LDS_STRIDE];

  const int m0 = blockIdx.x * 16;

#if USE_TDM
  if (threadIdx.x < 32) {   // one wave issues the DMA; TDM ignores EXEC
    const unsigned long long ga =
        (unsigned long long)(uintptr_t)(A + (unsigned long long)m0 * H);
    const unsigned lds_off = (unsigned)(uintptr_t)As;  // addr[31:0] = LDS byte offset
    const unsigned rows    = (unsigned)(M - m0);       // tensor_dim1 (OOB rows -> 0)

    u32x4 g0;
    g0[0] = 1u;                                        // count=1, user mode, no gather
    g0[1] = lds_off;                                   // lds_addr           [63:32]
    g0[2] = (unsigned)ga;                              // global_addr[31:0]  [95:64]
    g0[3] = (unsigned)((ga >> 32) & 0x01FFFFFFull)     // global_addr[56:32]
          | (2u << 30);                                // type=2 ("image")   [127:126]

    i32x8 g1 = (i32x8)0;
    g1[0] = (2 << 16)     // data_size = 4 bytes
          | (1 << 20)     // pad_enable
          | (6 << 22)     // pad_interval: 128 DWORDs
          | (3 << 25);    // pad_amount:   4 DWORDs  -> LDS row stride 132 floats
    g1[1] = (int)((128u & 0xFFFFu) << 16);                         // tensor_dim0 lo16 [63:48]
    g1[2] = (int)(((128u >> 16) & 0xFFFFu) | ((rows & 0xFFFFu) << 16)); // dim0 hi | dim1 lo
    g1[3] = (int)(((rows >> 16) & 0xFFFFu) | (128u << 16));        // dim1 hi | tile_dim0=128
    g1[4] = 16;            // tile_dim1 = 16 rows (tile_dim2 = 0)
    g1[5] = 128;           // tensor_dim0_stride = 128 elements (lo32)
                           // g1[6], g1[7] = 0 (stride hi, dim1_stride unused for 2D tile)

    i32x4 g2 = (i32x4)0;   // 2D tensor: groups 2/3 unused (zero-filled)
    i32x4 g3 = (i32x4)0;

#if __clang_major__ >= 23
    i32x8 g4 = (i32x8)0;   // clang-23 / therock HIP headers: 6-arg form
    __builtin_amdgcn_tensor_load_to_lds(g0, g1, g2, g3, g4, 0);
#else
    __builtin_amdgcn_tensor_load_to_lds(g0, g1, g2, g3, 0);        // ROCm 7.2 / clang-22
#endif
    __builtin_amdgcn_s_wait_tensorcnt(0);
  }
  __syncthreads();
#else
  // Fallback: cooperative, coalesced staging of the 16x128 A tile
  for (int i = threadIdx.x; i < 16 * H; i += 256) {
    int r = i >> 7, c = i & 127;
    As[r * LDS_STRIDE + c] = (m0 + r < M) ? A[(long long)(m0 + r) * H + c] : 0.0f;
  }
  __syncthreads();
#endif

  const int wave    = threadIdx.x >> 5;     // 0..7  -> N tile
  const int lane    = threadIdx.x & 31;
  const int laneRow = lane & 15;
  const int hi      = lane >> 4;            // 0: K+0/1, 1: K+2/3 (f32 16x16x4 layout)
  const int n0      = wave * 16;
  const int col     = n0 + laneRow;

  const float* Ws[4] = {W0, W1, W2, W3};
  const float* Bs[4] = {b0, b1, b2, b3};
  float*       Os[4] = {O0, O1, O2, O3};

#pragma unroll
  for (int wi = 0; wi < 4; ++wi) {
    const float* __restrict__ W = Ws[wi];
    v8f acc = {};
#pragma unroll
    for (int k = 0; k < H; k += 4) {
      // A fragment: lanes 0-15 rows 0-15 (K=k,k+1); lanes 16-31 rows 0-15 (K=k+2,k+3)
      v2f af;
      af.x = As[laneRow * LDS_STRIDE + k + 2 * hi];
      af.y = As[laneRow * LDS_STRIDE + k + 2 * hi + 1];
      // B fragment (K x N, N striped across lanes)
      v2f bf;
      bf.x = W[(k + 2 * hi) * H + col];
      bf.y = W[(k + 2 * hi + 1) * H + col];
      acc = __builtin_amdgcn_wmma_f32_16x16x4_f32(false, af, false, bf,
                                                  (short)0, acc, false, false);
    }
    const float bias = Bs[wi][col];
    float* __restrict__ O = Os[wi];
#pragma unroll
    for (int j = 0; j < 8; ++j) {
      int row = m0 + j + hi * 8;
      if (row < M) O[(long long)row * H + col] = acc[j] + bias;
    }
  }
}

// =====================================================================
// Per-layer init: acc = 0, den = 0, m_enc = enc(-inf)
// =====================================================================
__global__ __launch_bounds__(256) void init_kernel(float* __restrict__ acc,
                                                   float* __restrict__ den,
                                                   unsigned* __restrict__ m_enc,
                                                   int N)
{
  long long i = (long long)blockIdx.x * blockDim.x + threadIdx.x;
  if (i < (long long)N * H) acc[i] = 0.0f;
  if (i < N) { den[i] = 0.0f; m_enc[i] = 0x007FFFFFu; /* enc(-inf) */ }
}

// =====================================================================
// Edge pass A: alpha_e = dot(q[dst], k[src] (+ edge_attr@Ew + Eb)) / sqrt(H)
//              segment-max over dst via ordered-uint atomicMax.
// One wave per edge; lane owns 4 channels; edge projection fused:
// Ew (32x128) staged in LDS, edge_attr broadcast across the wave via __shfl.
// =====================================================================
template <bool HAS_EF>
__global__ __launch_bounds__(256) void edge_alpha_kernel(
    const long long* __restrict__ ei, int E,
    const float* __restrict__ edge_attr,
    const float* __restrict__ Ew, const float* __restrict__ Eb,
    const float* __restrict__ Q, const float* __restrict__ K,
    unsigned* __restrict__ m_enc, float* __restrict__ alpha)
{
  __shared__ float ews[HAS_EF ? ED * H : 1];
  __shared__ float ebs[HAS_EF ? H : 1];
  if (HAS_EF) {
    for (int i = threadIdx.x; i < ED * H; i += 256) ews[i] = Ew[i];
    for (int i = threadIdx.x; i < H; i += 256) ebs[i] = Eb[i];
    __syncthreads();
  }
  const int lane = threadIdx.x & 31;
  const int e = blockIdx.x * 8 + (threadIdx.x >> 5);
  if (e >= E) return;

  const int src = (int)ei[e];
  const int dst = (int)ei[E + e];
  const int c0 = lane * 4;

  float4 kv = *(const float4*)(K + (long long)src * H + c0);
  float4 qv = *(const float4*)(Q + (long long)dst * H + c0);

  float ef0 = 0.f, ef1 = 0.f, ef2 = 0.f, ef3 = 0.f;
  if (HAS_EF) {
    float my_ea = edge_attr[(long long)e * ED + lane];   // ED == 32 == wave width
    float4 bb = *(const float4*)(ebs + c0);
    ef0 = bb.x; ef1 = bb.y; ef2 = bb.z; ef3 = bb.w;
#pragma unroll
    for (int j = 0; j < ED; ++j) {
      float a = __shfl(my_ea, j, 32);
      float4 wv = *(const float4*)(ews + j * H + c0);
      ef0 += a * wv.x; ef1 += a * wv.y; ef2 += a * wv.z; ef3 += a * wv.w;
    }
  }
  float p = qv.x * (kv.x + ef0) + qv.y * (kv.y + ef1) +
            qv.z * (kv.z + ef2) + qv.w * (kv.w + ef3);
#pragma unroll
  for (int off = 16; off >= 1; off >>= 1) p += __shfl_xor(p, off, 32);

  if (lane == 0) {
    float a = p * 0.08838834764831845f;  // 1/sqrt(128)
    alpha[e] = a;
    atomicMax(m_enc + dst, fenc(a));
  }
}

// =====================================================================
// Edge pass B: ex = exp(alpha - m[dst]); den[dst] += ex;
//              acc[dst] += ex * (v[src] (+ edge_feat))
// Normalization by den is deferred to the per-node finalize pass
// (softmax: out = (sum ex*v) / (sum ex)).
// =====================================================================
template <bool HAS_EF>
__global__ __launch_bounds__(256) void edge_scatter_kernel(
    const long long* __restrict__ ei, int E,
    const float* __restrict__ edge_attr,
    const float* __restrict__ Ew, const float* __restrict__ Eb,
    const float* __restrict__ V, const float* __restrict__ alpha,
    const unsigned* __restrict__ m_enc,
    float* __restrict__ den, float* __restrict__ acc)
{
  __shared__ float ews[HAS_EF ? ED * H : 1];
  __shared__ float ebs[HAS_EF ? H : 1];
  if (HAS_EF) {
    for (int i = threadIdx.x; i < ED * H; i += 256) ews[i] = Ew[i];
    for (int i = threadIdx.x; i < H; i += 256) ebs[i] = Eb[i];
    __syncthreads();
  }
  const int lane = threadIdx.x & 31;
  const int e = blockIdx.x * 8 + (threadIdx.x >> 5);
  if (e >= E) return;

  const int src = (int)ei[e];
  const int dst = (int)ei[E + e];
  const int c0 = lane * 4;

  float m  = fdec(m_enc[dst]);
  float ex = __expf(alpha[e] - m);
  if (lane == 0) atomAddF(den + dst, ex);

  float ef0 = 0.f, ef1 = 0.f, ef2 = 0.f, ef3 = 0.f;
  if (HAS_EF) {
    float my_ea = edge_attr[(long long)e * ED + lane];
    float4 bb = *(const float4*)(ebs + c0);
    ef0 = bb.x; ef1 = bb.y; ef2 = bb.z; ef3 = bb.w;
#pragma unroll
    for (int j = 0; j < ED; ++j) {
      float a = __shfl(my_ea, j, 32);
      float4 wv = *(const float4*)(ews + j * H + c0);
      ef0 += a * wv.x; ef1 += a * wv.y; ef2 += a * wv.z; ef3 += a * wv.w;
    }
  }
  float4 vv = *(const float4*)(V + (long long)src * H + c0);
  float* ap = acc + (long long)dst * H + c0;
  atomAddF(ap + 0, ex * (vv.x + ef0));
  atomAddF(ap + 1, ex * (vv.y + ef1));
  atomAddF(ap + 2, ex * (vv.z + ef2));
  atomAddF(ap + 3, ex * (vv.w + ef3));
}

// =====================================================================
// Finalize: out = acc/(den+1e-16) + skip ; optional ReLU (layer 1)
// Empty segments: acc=0, den=0 -> out = skip, matching the reference.
// =====================================================================
template <bool RELU>
__global__ __launch_bounds__(256) void finalize_kernel(
    const float* __restrict__ acc, const float* __restrict__ den,
    const float* __restrict__ skip, float* __restrict__ out, int N)
{
  long long i = (long long)blockIdx.x * blockDim.x + threadIdx.x;
  if (i >= (long long)N * H) return;
  int n = (int)(i >> 7);
  float v = acc[i] / (den[n] + 1e-16f) + skip[i];
  out[i] = RELU ? fmaxf(v, 0.0f) : v;
}

// =====================================================================
// Host launcher
// =====================================================================
extern "C" void kernel_launch(void* const* d_in, const int* in_sizes, int n_in,
                              void* d_out, int out_size, void* d_ws, size_t ws_size,
                              hipStream_t stream)
{
  // Inputs in setup_inputs() order
  const long long* ei  = (const long long*)d_in[0];   // int64 (2,E)
  const float* edge_attr = (const float*)d_in[1];     // (E,32)
  const float* emb  = (const float*)d_in[2];          // (N,128)
  const float* q1_w = (const float*)d_in[3];  const float* q1_b = (const float*)d_in[4];
  const float* k1_w = (const float*)d_in[5];  const float* k1_b = (const float*)d_in[6];
  const float* v1_w = (const float*)d_in[7];  const float* v1_b = (const float*)d_in[8];
  const float* e1_w = (const float*)d_in[9];  const float* e1_b = (const float*)d_in[10];
  const float* s1_w = (const float*)d_in[11]; const float* s1_b = (const float*)d_in[12];
  const float* q2_w = (const float*)d_in[13]; const float* q2_b = (const float*)d_in[14];
  const float* k2_w = (const float*)d_in[15]; const float* k2_b = (const float*)d_in[16];
  const float* v2_w = (const float*)d_in[17]; const float* v2_b = (const float*)d_in[18];
  const float* s2_w = (const float*)d_in[19]; const float* s2_b = (const float*)d_in[20];

  const int E = in_sizes[0] / 2;
  const int N = in_sizes[2] / H;
  const long long NH = (long long)N * H;

  // Workspace carve-up (~158 MB for N=50k, E=800k)
  float* Q     = (float*)d_ws;
  float* Kb    = Q + NH;
  float* Vb    = Kb + NH;
  float* Sb    = Vb + NH;
  float* ACC   = Sb + NH;
  float* X1    = ACC + NH;
  float* ALPHA = X1 + NH;           // E floats
  float* DEN   = ALPHA + E;         // N floats
  unsigned* MENC = (unsigned*)(DEN + N);  // N uints

  float* OUT = (float*)d_out;

  const int gemmBlocks = (N + 15) / 16;
  const int nhBlocks   = (int)((NH + 255) / 256);
  const int edgeBlocks = (E + 7) / 8;        // 8 waves (edges) per 256-thread block

  // ---------------- Layer 1 ----------------
  gemm4_kernel<<<gemmBlocks, 256, 0, stream>>>(
      emb, N,
      q1_w, q1_b, Q,  k1_w, k1_b, Kb,  v1_w, v1_b, Vb,  s1_w, s1_b, Sb);
  init_kernel<<<nhBlocks, 256, 0, stream>>>(ACC, DEN, MENC, N);
  edge_alpha_kernel<true><<<edgeBlocks, 256, 0, stream>>>(
      ei, E, edge_attr, e1_w, e1_b, Q, Kb, MENC, ALPHA);
  edge_scatter_kernel<true><<<edgeBlocks, 256, 0, stream>>>(
      ei, E, edge_attr, e1_w, e1_b, Vb, ALPHA, MENC, DEN, ACC);
  finalize_kernel<true><<<nhBlocks, 256, 0, stream>>>(ACC, DEN, Sb, X1, N);

  // ---------------- Layer 2 ----------------
  gemm4_kernel<<<gemmBlocks, 256, 0, stream>>>(
      X1, N,
      q2_w, q2_b, Q,  k2_w, k2_b, Kb,  v2_w, v2_b, Vb,  s2_w, s2_b, Sb);
  init_kernel<<<nhBlocks, 256, 0, stream>>>(ACC, DEN, MENC, N);
  edge_alpha_kernel<false><<<edgeBlocks, 256, 0, stream>>>(
      ei, E, nullptr, nullptr, nullptr, Q, Kb, MENC, ALPHA);
  edge_scatter_kernel<false><<<edgeBlocks, 256, 0, stream>>>(
      ei, E, nullptr, nullptr, nullptr, Vb, ALPHA, MENC, DEN, ACC);
  finalize_kernel<false><<<nhBlocks, 256, 0, stream>>>(ACC, DEN, Sb, OUT, N);
}